// QwenMoEGate_50964081935156
// MI455X (gfx1250) — compile-verified
//
#include <hip/hip_runtime.h>
#include <hip/hip_bf16.h>

typedef __attribute__((ext_vector_type(2))) float v2f;
typedef __attribute__((ext_vector_type(4))) float v4f;
typedef __attribute__((ext_vector_type(8))) float v8f;

#define HIDDEN 4096
#define NEXP 64
#define TOPK 8
#define TOKENS 16384
#define KC 512                 // K-chunk staged in LDS
#define NCHUNK (HIDDEN / KC)   // 8
#define WROW (KC + 4)          // padded row stride (dwords): conflict-free b64 lane pattern
#define TOK_PER_BLOCK 128      // 8 wave32 waves x 16 tokens
#define THREADS 256
#define LSTRIDE 68             // logits row stride (dwords)

union Smem {
  float w[NEXP * WROW];        // 64 * 516 * 4 = 132096 bytes (weight chunk)
  struct {
    float logits[TOK_PER_BLOCK * LSTRIDE];
    float psum[NEXP];
    float pcnt[NEXP];
  } s;
};

__global__ __launch_bounds__(THREADS) void moe_gate_main(
    const float* __restrict__ x, const float* __restrict__ w,
    int* __restrict__ outIdx, float* __restrict__ outW,
    float* __restrict__ gSum, float* __restrict__ gCnt) {
  __shared__ Smem sm;
  const int tid  = threadIdx.x;
  const int lane = tid & 31;
  const int wave = tid >> 5;
  const int tok0 = blockIdx.x * TOK_PER_BLOCK + wave * 16;

  v8f acc[4];
#pragma unroll
  for (int t = 0; t < 4; ++t) acc[t] = v8f{0.f, 0.f, 0.f, 0.f, 0.f, 0.f, 0.f, 0.f};

  // A-fragment base: lanes 0-15 -> rows M=0..15 at K, lanes 16-31 -> same rows at K+2
  const float* aBase = x + (size_t)(tok0 + (lane & 15)) * HIDDEN + ((lane >> 4) * 2);

  for (int c = 0; c < NCHUNK; ++c) {
    __syncthreads();   // previous chunk's LDS reads complete before overwrite
    const int k0 = c * KC;
    // cooperative stage: weight[0..63][k0 .. k0+KC) -> LDS, b128 per thread-iter
#pragma unroll 1
    for (int i = tid; i < NEXP * (KC / 4); i += THREADS) {
      const int e = i >> 7;           // KC/4 == 128 quads per expert row
      const int q = (i & 127) << 2;   // dword column within chunk
      const v4f vw = *(const v4f*)(w + (size_t)e * HIDDEN + k0 + q);
      *(v4f*)(&sm.w[e * WROW + q]) = vw;
    }
    __syncthreads();

    const float* ap = aBase + k0;
#pragma unroll 4
    for (int kk = 0; kk < KC; kk += 4) {
      const v2f a = *(const v2f*)(ap + kk);   // global b64: (K,K+1)/(K+2,K+3) per lane half
#pragma unroll
      for (int t = 0; t < 4; ++t) {
        const int e = t * 16 + (lane & 15);
        const v2f b = *(const v2f*)(&sm.w[e * WROW + kk + ((lane >> 4) * 2)]); // ds b64
        acc[t] = __builtin_amdgcn_wmma_f32_16x16x4_f32(
            false, a, false, b, (short)0, acc[t], false, false);
      }
    }
  }

  // ---- spill logits (16 tokens x 64 experts per wave) to LDS ----
  __syncthreads();   // all waves done reading sm.w before aliasing it with logits
#pragma unroll
  for (int t = 0; t < 4; ++t)
#pragma unroll
    for (int j = 0; j < 8; ++j) {
      const int tl = wave * 16 + ((lane < 16) ? j : j + 8);   // C layout: VGPR j = M=j / M=j+8
      sm.s.logits[tl * LSTRIDE + t * 16 + (lane & 15)] = acc[t][j];
    }
  if (tid < NEXP) { sm.s.psum[tid] = 0.f; sm.s.pcnt[tid] = 0.f; }
  __syncthreads();

  // ---- per-token softmax + top-8 + aux contributions ----
  if (tid < TOK_PER_BLOCK) {
    const float* row = &sm.s.logits[tid * LSTRIDE];
    float mx = -3.4e38f;
    for (int e = 0; e < NEXP; ++e) mx = fmaxf(mx, row[e]);
    float se = 0.f;
    for (int e = 0; e < NEXP; ++e) se += __expf(row[e] - mx);
    const float inv = 1.f / se;

    const int gtok = blockIdx.x * TOK_PER_BLOCK + tid;
    unsigned long long taken = 0ull;
    float tw[TOPK];
    float tsum = 0.f;
    int first = 0;
#pragma unroll 1
    for (int j = 0; j < TOPK; ++j) {
      float best = -3.4e38f;
      int bi = 0;
      for (int e = 0; e < NEXP; ++e) {
        const float v = row[e];
        if (!((taken >> e) & 1ull) && v > best) { best = v; bi = e; }  // strict > : lowest index on ties
      }
      taken |= (1ull << bi);
      if (j == 0) first = bi;
      const float p = __expf(best - mx) * inv;
      tw[j] = p;
      tsum += p;
      outIdx[gtok * TOPK + j] = bi;
    }
    const float r = 1.f / (tsum + 1e-20f);
#pragma unroll
    for (int j = 0; j < TOPK; ++j) outW[gtok * TOPK + j] = tw[j] * r;

    atomicAdd(&sm.s.pcnt[first], 1.f);                       // argmax histogram (LDS)
    for (int e = 0; e < NEXP; ++e)
      atomicAdd(&sm.s.psum[e], __expf(row[e] - mx) * inv);   // prob sums (LDS)
  }
  __syncthreads();
  if (tid < NEXP) {
    atomicAdd(&gSum[tid], sm.s.psum[tid]);
    atomicAdd(&gCnt[tid], sm.s.pcnt[tid]);
  }
}

__global__ void moe_gate_zero(float* __restrict__ ws) {
  if (threadIdx.x < 2 * NEXP) ws[threadIdx.x] = 0.f;
}

__global__ void moe_gate_aux(const float* __restrict__ gSum,
                             const float* __restrict__ gCnt,
                             float* __restrict__ aux) {
  float a = 0.f;
  for (int e = 0; e < NEXP; ++e)
    a += (gSum[e] / (float)TOKENS) * (gCnt[e] / (float)TOKENS);
  *aux = a * (float)NEXP;
}

extern "C" void kernel_launch(void* const* d_in, const int* in_sizes, int n_in,
                              void* d_out, int out_size, void* d_ws, size_t ws_size,
                              hipStream_t stream) {
  const float* x = (const float*)d_in[0];    // [4,4096,4096] f32
  const float* w = (const float*)d_in[1];    // [64,4096] f32

  int*   outIdx = (int*)d_out;                             // 16384*8 int32
  float* outW   = (float*)d_out + TOKENS * TOPK;           // 16384*8 f32
  float* outAux = (float*)d_out + 2 * TOKENS * TOPK;       // 1 f32

  float* gSum = (float*)d_ws;        // [64] mean-prob accumulators
  float* gCnt = gSum + NEXP;         // [64] argmax counts

  moe_gate_zero<<<1, 128, 0, stream>>>(gSum);
  moe_gate_main<<<TOKENS / TOK_PER_BLOCK, THREADS, 0, stream>>>(
      x, w, outIdx, outW, gSum, gCnt);
  moe_gate_aux<<<1, 1, 0, stream>>>(gSum, gCnt, outAux);
}